// ProbAttention_37538014167159
// MI455X (gfx1250) — compile-verified
//
#include <hip/hip_runtime.h>

typedef __attribute__((ext_vector_type(2))) float v2f;
typedef __attribute__((ext_vector_type(8))) float v8f;

#define WMMA_F32(a, b, c) \
  __builtin_amdgcn_wmma_f32_16x16x4_f32(false, (a), false, (b), (short)0, (c), false, false)

constexpr int Bsz  = 4;
constexpr int Lk   = 2048;
constexpr int Hn   = 8;
constexpr int Dd   = 64;
constexpr int NTOP = 128;   // SAMPLE_K == n_top == 128

// ---------------------------------------------------------------------------
// Kernel 1: sparsity measure M[b,h,q] = max_s(QK_s) - sum_s(QK_s)/L
// grid = B*H*L blocks of 128 threads (one thread per sampled key)
// ---------------------------------------------------------------------------
__global__ __launch_bounds__(128) void pa_measure(
    const float* __restrict__ q, const float* __restrict__ k,
    const int* __restrict__ idx, float* __restrict__ Mout)
{
  __shared__ __align__(16) float sQ[Dd];
  __shared__ float smax[128];
  __shared__ float ssum[128];

  const int blk  = blockIdx.x;          // bh*Lk + qpos
  const int qpos = blk & (Lk - 1);
  const int bh   = blk >> 11;
  const int b    = bh >> 3;
  const int h    = bh & 7;
  const int t    = threadIdx.x;

  if (t < Dd) sQ[t] = q[(((size_t)(b * Lk + qpos)) * Hn + h) * Dd + t];
  __syncthreads();

  const int key = idx[qpos * NTOP + t];
  const float4* krow = (const float4*)(k + (((size_t)(b * Lk + key)) * Hn + h) * Dd);
  const float4* q4   = (const float4*)sQ;
  float dot = 0.f;
  #pragma unroll
  for (int i = 0; i < 16; ++i) {
    float4 a = q4[i], bb = krow[i];
    dot += a.x * bb.x + a.y * bb.y + a.z * bb.z + a.w * bb.w;
  }
  smax[t] = dot;
  ssum[t] = dot;
  for (int s2 = 64; s2 > 0; s2 >>= 1) {
    __syncthreads();
    if (t < s2) {
      smax[t] = fmaxf(smax[t], smax[t + s2]);
      ssum[t] += ssum[t + s2];
    }
  }
  __syncthreads();
  if (t == 0) Mout[blk] = smax[0] - ssum[0] * (1.0f / (float)Lk);
}

// ---------------------------------------------------------------------------
// Kernel 2: exact top-128 per (b,h) by rank counting (matches top_k tie order)
// grid = B*H blocks of 256 threads
// ---------------------------------------------------------------------------
__global__ __launch_bounds__(256) void pa_topk(
    const float* __restrict__ M, int* __restrict__ mtop)
{
  __shared__ float sM[Lk];
  const int bh = blockIdx.x;
  const int t  = threadIdx.x;
  for (int i = t; i < Lk; i += 256) sM[i] = M[(size_t)bh * Lk + i];
  __syncthreads();
  #pragma unroll
  for (int i = 0; i < 8; ++i) {
    const int qq = t * 8 + i;
    const float mi = sM[qq];
    int rank = 0;
    for (int j = 0; j < Lk; ++j) {
      const float mj = sM[j];
      rank += (mj > mi) || (mj == mi && j < qq);
    }
    if (rank < NTOP) mtop[bh * NTOP + rank] = qq;
  }
}

// ---------------------------------------------------------------------------
// Kernels 3/4: chunked prefix sum of V along L -> out  (out[b,l,h,d])
// grid = B*H*16 blocks of 64 threads (one thread per d), chunk = 128 rows
// ---------------------------------------------------------------------------
__global__ __launch_bounds__(64) void pa_cumsum_local(
    const float* __restrict__ v, float* __restrict__ out, float* __restrict__ csum)
{
  const int blk = blockIdx.x;
  const int c   = blk & 15;
  const int bh  = blk >> 4;
  const int b   = bh >> 3;
  const int h   = bh & 7;
  const int d   = threadIdx.x;
  float run = 0.f;
  const int j0 = c * 128;
  for (int j = j0; j < j0 + 128; ++j) {
    const size_t off = (((size_t)(b * Lk + j)) * Hn + h) * Dd + d;
    run += v[off];
    out[off] = run;
  }
  csum[((size_t)bh * 16 + c) * Dd + d] = run;
}

__global__ __launch_bounds__(64) void pa_cumsum_fix(
    float* __restrict__ out, const float* __restrict__ csum)
{
  const int blk = blockIdx.x;
  const int c   = blk & 15;
  if (c == 0) return;
  const int bh  = blk >> 4;
  const int b   = bh >> 3;
  const int h   = bh & 7;
  const int d   = threadIdx.x;
  float offv = 0.f;
  for (int cc = 0; cc < c; ++cc) offv += csum[((size_t)bh * 16 + cc) * Dd + d];
  const int j0 = c * 128;
  for (int j = j0; j < j0 + 128; ++j) {
    const size_t off = (((size_t)(b * Lk + j)) * Hn + h) * Dd + d;
    out[off] += offv;
  }
}

// ---------------------------------------------------------------------------
// Kernel 5: flash-style masked attention over top queries using f32 WMMA.
//   scores[u,k] = ((Qr[u]+u_w[u])·K[k] + (Qr[u]+v_w[u])·Kp[k] + u_b[u]+v_b[u]) * 0.125
//   masked where k > pos[u]; online softmax; acc = attn @ V; scatter to out.
// grid = B*H*4 blocks of 64 threads (2 waves, 16 query-rows per wave)
// ---------------------------------------------------------------------------
__global__ __launch_bounds__(64) void pa_core(
    const float* __restrict__ q,   const float* __restrict__ kmat,
    const float* __restrict__ kpm, const float* __restrict__ vmat,
    const float* __restrict__ u_w, const float* __restrict__ u_b,
    const float* __restrict__ v_w, const float* __restrict__ v_b,
    const int*  __restrict__ mtop, float* __restrict__ out)
{
  __shared__ __align__(16) float sA1[32 * 64];
  __shared__ __align__(16) float sA2[32 * 64];
  __shared__ __align__(16) float sK [32 * 64];
  __shared__ __align__(16) float sKp[32 * 64];
  __shared__ __align__(16) float sV [32 * 64];
  __shared__ __align__(16) float sP [2][16 * 16];
  __shared__ float sBias[32];
  __shared__ int   sPos[32];

  const int blk   = blockIdx.x;
  const int split = blk & 3;          // 4 splits x 32 rows = 128 top queries
  const int bh    = blk >> 2;
  const int b     = bh >> 3;
  const int h     = bh & 7;
  const int tid   = threadIdx.x;      // 64 threads = 2 waves
  const int wave  = tid >> 5;
  const int lane  = tid & 31;
  const int hi    = lane >> 4;        // half-wave select
  const int ln    = lane & 15;
  const int koff  = hi * 2;           // K-subcolumn offset for f32 A/B frags

  // ---- setup: A1 = Q[pos]+u_w, A2 = Q[pos]+v_w, bias, positions ----
  {
    const int u_loc  = tid >> 1;
    const int half   = tid & 1;
    const int u_glob = split * 32 + u_loc;
    const int pos    = mtop[bh * NTOP + u_glob];
    if (half == 0) {
      sPos[u_loc]  = pos;
      sBias[u_loc] = u_b[u_glob] + v_b[u_glob];
    }
    const float* qrow = q   + (((size_t)(b * Lk + pos)) * Hn + h) * Dd;
    const float* uwr  = u_w + (size_t)u_glob * Dd;
    const float* vwr  = v_w + (size_t)u_glob * Dd;
    #pragma unroll
    for (int i = 0; i < 32; ++i) {
      const int d  = half * 32 + i;
      const float qv = qrow[d];
      sA1[u_loc * 64 + d] = qv + uwr[d];
      sA2[u_loc * 64 + d] = qv + vwr[d];
    }
  }

  float m_st[8], l_st[8];
  v8f acc[4];
  #pragma unroll
  for (int r = 0; r < 8; ++r) { m_st[r] = -__builtin_inff(); l_st[r] = 0.f; }
  #pragma unroll
  for (int dn = 0; dn < 4; ++dn) acc[dn] = {};

  for (int kt = 0; kt < Lk; kt += 32) {
    __syncthreads();
    // stage K/Kp/V tiles: 32 keys x 64 floats each, float4 coalesced
    #pragma unroll
    for (int i = 0; i < 8; ++i) {
      const int f   = tid + i * 64;       // 0..511 float4 slots
      const int key = f >> 4;
      const int dv  = f & 15;
      const size_t rb = (((size_t)(b * Lk + kt + key)) * Hn + h) * 16;  // float4 units
      ((float4*)sK )[key * 16 + dv] = ((const float4*)kmat)[rb + dv];
      ((float4*)sKp)[key * 16 + dv] = ((const float4*)kpm )[rb + dv];
      ((float4*)sV )[key * 16 + dv] = ((const float4*)vmat)[rb + dv];
    }
    __syncthreads();

    #pragma unroll
    for (int ks = 0; ks < 2; ++ks) {
      // ---- S(16x16) = A1 @ K^T + A2 @ Kp^T via 32 chained f32 WMMAs ----
      v8f c = {};
      #pragma unroll
      for (int kk = 0; kk < 16; ++kk) {
        v2f a  = *(const v2f*)&sA1[(wave * 16 + ln) * 64 + kk * 4 + koff];
        v2f bb = *(const v2f*)&sK [(ks   * 16 + ln) * 64 + kk * 4 + koff];
        c = WMMA_F32(a, bb, c);
      }
      #pragma unroll
      for (int kk = 0; kk < 16; ++kk) {
        v2f a  = *(const v2f*)&sA2[(wave * 16 + ln) * 64 + kk * 4 + koff];
        v2f bb = *(const v2f*)&sKp[(ks   * 16 + ln) * 64 + kk * 4 + koff];
        c = WMMA_F32(a, bb, c);
      }

      // ---- online softmax (row = wave*16 + hi*8 + r, cols across 16 lanes) ----
      const int kg = kt + ks * 16 + ln;
      #pragma unroll
      for (int r = 0; r < 8; ++r) {
        const int u_loc = wave * 16 + hi * 8 + r;
        float s = (c[r] + sBias[u_loc]) * 0.125f;
        if (kg > sPos[u_loc]) s = -__builtin_inff();
        float rm = s;
        rm = fmaxf(rm, __shfl_xor(rm, 1));
        rm = fmaxf(rm, __shfl_xor(rm, 2));
        rm = fmaxf(rm, __shfl_xor(rm, 4));
        rm = fmaxf(rm, __shfl_xor(rm, 8));
        const float m_new = fmaxf(m_st[r], rm);
        float p = (m_new == -__builtin_inff()) ? 0.f : expf(s - m_new);
        float rs = p;
        rs += __shfl_xor(rs, 1);
        rs += __shfl_xor(rs, 2);
        rs += __shfl_xor(rs, 4);
        rs += __shfl_xor(rs, 8);
        const float sc = (m_st[r] == -__builtin_inff()) ? 0.f : expf(m_st[r] - m_new);
        l_st[r] = l_st[r] * sc + rs;
        m_st[r] = m_new;
        #pragma unroll
        for (int dn = 0; dn < 4; ++dn) acc[dn][r] *= sc;
        sP[wave][(hi * 8 + r) * 16 + ln] = p;   // stage P for A-frag transpose
      }

      // ---- acc += P @ V_sub (wave-private sP; same-wave LDS order suffices) ----
      #pragma unroll
      for (int dn = 0; dn < 4; ++dn) {
        #pragma unroll
        for (int kk = 0; kk < 4; ++kk) {
          v2f a = *(const v2f*)&sP[wave][ln * 16 + kk * 4 + koff];
          v2f bb;
          bb[0] = sV[(ks * 16 + kk * 4 + koff    ) * 64 + dn * 16 + ln];
          bb[1] = sV[(ks * 16 + kk * 4 + koff + 1) * 64 + dn * 16 + ln];
          acc[dn] = WMMA_F32(a, bb, acc[dn]);
        }
      }
    }
  }

  // ---- normalize and scatter rows over the cumsum baseline ----
  #pragma unroll
  for (int r = 0; r < 8; ++r) {
    const int u_loc = wave * 16 + hi * 8 + r;
    const int pos   = sPos[u_loc];
    const float inv = 1.0f / l_st[r];
    const size_t ob = (((size_t)(b * Lk + pos)) * Hn + h) * Dd;
    #pragma unroll
    for (int dn = 0; dn < 4; ++dn)
      out[ob + dn * 16 + ln] = acc[dn][r] * inv;
  }
}

// ---------------------------------------------------------------------------
extern "C" void kernel_launch(void* const* d_in, const int* in_sizes, int n_in,
                              void* d_out, int out_size, void* d_ws, size_t ws_size,
                              hipStream_t stream) {
  const float* q   = (const float*)d_in[0];   // queries (B,L,H,D)
  const float* k   = (const float*)d_in[1];   // keys
  const float* v   = (const float*)d_in[2];   // values
  const float* kp  = (const float*)d_in[3];   // position_embedding_key
  const float* u_w = (const float*)d_in[4];   // (128,64)
  const float* u_b = (const float*)d_in[5];   // (128,)
  const float* v_w = (const float*)d_in[6];
  const float* v_b = (const float*)d_in[7];
  const int*   idx = (const int*)d_in[8];     // (L,128)
  float* out = (float*)d_out;

  char* ws = (char*)d_ws;
  float* Mbuf = (float*)ws;                                   // B*H*L   = 65536 f
  int*   mtop = (int*)(ws + (size_t)65536 * 4);               // B*H*128 = 4096 i
  float* csum = (float*)(ws + (size_t)65536 * 4 + 4096 * 4);  // B*H*16*64 f

  pa_measure     <<<Bsz * Hn * Lk, 128, 0, stream>>>(q, k, idx, Mbuf);
  pa_topk        <<<Bsz * Hn,      256, 0, stream>>>(Mbuf, mtop);
  pa_cumsum_local<<<Bsz * Hn * 16,  64, 0, stream>>>(v, out, csum);
  pa_cumsum_fix  <<<Bsz * Hn * 16,  64, 0, stream>>>(out, csum);
  pa_core        <<<Bsz * Hn * 4,   64, 0, stream>>>(q, k, kp, v, u_w, u_b, v_w, v_b, mtop, out);
}